// GaussianFalloffKernel_83434034692733
// MI455X (gfx1250) — compile-verified
//
#include <hip/hip_runtime.h>

typedef __attribute__((ext_vector_type(2))) float v2f;
typedef __attribute__((ext_vector_type(8))) float v8f;

#define NVORT 512
#define NPTS  65536   // 256*256 points per batch
#define LOG2E 1.4426950408889634f

// Each wave (32 lanes, wave32) computes a 16-point tile against all 512 vortices.
// sq distances for a 16x16 (point x vortex) tile come from one
// V_WMMA_F32_16X16X4_F32:  D = A(16x4)[p0 p1 0 0] * B(4x16)[-2y;-2x;0;0] + C(|p|^2+|l|^2)
// D layout: lane = N + 16*(M>=8), VGPR v = M%8  -> elementwise exp/rsq work per lane.
// Per-chunk LDS reads are software-pipelined one iteration ahead so the
// s_wait_dscnt is hidden behind the ~70-op elementwise body.
__global__ __launch_bounds__(256) void vortex_vel_kernel(
    const float* __restrict__ vf,      // (2, 512, 6): y,x,tau,sig,v,u
    const float* __restrict__ pts_raw, // (2, 65536, 2)
    float* __restrict__ out)           // (2, 65536, 2)
{
    __shared__ float sY[NVORT], sX[NVORT], sC[NVORT], sT[NVORT], sL[NVORT];

    const int tid = threadIdx.x;
    const int b   = blockIdx.x >> 9;          // 512 blocks per batch

    // Stage per-vortex derived data into LDS once per block.
    for (int k = tid; k < NVORT; k += 256) {
        const float* f = vf + ((size_t)b * NVORT + k) * 6;
        float y = f[0], x = f[1], tau = f[2], sig = f[3];
        sY[k] = y;
        sX[k] = x;
        sT[k] = tau;
        sC[k] = -LOG2E / (sig * sig);         // exp(-sq/sig^2) = exp2(sq * sC)
        sL[k] = y * y + x * x;
    }
    __syncthreads();

    const int lane = tid & 31;
    const int half = lane >> 4;               // lane group: rows M=0..7 vs M=8..15
    const int wave = tid >> 5;
    const int tileBase = (((blockIdx.x & 511) << 3) + wave) << 4;   // 16 points/wave
    const float2* pts = (const float2*)pts_raw + (size_t)b * NPTS;

    // Per-lane row constants for the 8 C/D rows this lane sees (M = v + 8*half).
    float pr0[8], pr1[8], cb[8];
#pragma unroll
    for (int v = 0; v < 8; ++v) {
        float2 p = pts[tileBase + v + (half << 3)];
        pr0[v] = p.x;
        pr1[v] = p.y;
        cb[v]  = p.x * p.x + p.y * p.y;       // |p|^2
    }

    // A matrix 16x4: lanes 0-15 rows M=lane, VGPR0=K0(p0), VGPR1=K1(p1);
    // lanes 16-31 hold K2,K3 = 0.
    v2f A;
    {
        float2 p = pts[tileBase + (lane & 15)];
        A.x = half ? 0.0f : p.x;
        A.y = half ? 0.0f : p.y;
    }

    float acc0[8], acc1[8];                   // Σ s*dx , Σ s*dy (negated at store)
#pragma unroll
    for (int v = 0; v < 8; ++v) { acc0[v] = 0.0f; acc1[v] = 0.0f; }

    // Software-pipeline: preload chunk 0's per-vortex values.
    const int col = lane & 15;
    float ny = sY[col], nx = sX[col], cc = sC[col], tt = sT[col], ll = sL[col];

    for (int c = 0; c < NVORT / 16; ++c) {
        // Consume the values prefetched for this chunk.
        const float cny = ny, cnx = nx, ccc = cc, ctt = tt, cll = ll;

        // B matrix 4x16: VGPR0 = {K0 lanes0-15, K2 lanes16-31}, VGPR1 = {K1, K3}
        v2f B;
        B.x = half ? 0.0f : -2.0f * cny;
        B.y = half ? 0.0f : -2.0f * cnx;

        v8f C;
#pragma unroll
        for (int v = 0; v < 8; ++v) C[v] = cb[v] + cll;   // |p|^2 + |l|^2

        // D = sq distances for the 16x16 tile.
        v8f D = __builtin_amdgcn_wmma_f32_16x16x4_f32(
            false, A, false, B, (short)0, C, false, false);

        // Prefetch next chunk's LDS values while the WMMA result settles and
        // the elementwise body below runs (hides ds latency + wmma hazard).
        if (c + 1 < NVORT / 16) {
            const int n2 = ((c + 1) << 4) + col;
            ny = sY[n2]; nx = sX[n2]; cc = sC[n2]; tt = sT[n2]; ll = sL[n2];
        }

#pragma unroll
        for (int v = 0; v < 8; ++v) {
            float sq = D[v];
            float e  = __builtin_amdgcn_exp2f(sq * ccc);  // exp(-sq/sig^2)
            float r  = __builtin_amdgcn_rsqf(sq);         // 1/sqrt(sq)
            float s  = ctt * e * r;
            float dy = pr0[v] - cny;
            float dx = pr1[v] - cnx;
            acc0[v] = fmaf(s, dx, acc0[v]);
            acc1[v] = fmaf(s, dy, acc1[v]);
        }
    }

    // Reduce over the 16 vortex columns (lanes within each 16-lane half),
    // then 2 lanes per v store rows v and v+8.
#pragma unroll
    for (int v = 0; v < 8; ++v) {
        float s0 = acc0[v], s1 = acc1[v];
#pragma unroll
        for (int m = 1; m < 16; m <<= 1) {
            s0 += __shfl_xor(s0, m, 32);
            s1 += __shfl_xor(s1, m, 32);
        }
        if ((lane & 15) == v) {
            int row = v + (half << 3);
            float2* o = (float2*)out + (size_t)b * NPTS + tileBase + row;
            *o = make_float2(s0, -s1);        // out[...,0]=Σ s*dx, out[...,1]=-Σ s*dy
        }
    }
}

extern "C" void kernel_launch(void* const* d_in, const int* in_sizes, int n_in,
                              void* d_out, int out_size, void* d_ws, size_t ws_size,
                              hipStream_t stream) {
    const float* vf  = (const float*)d_in[0];   // (2,512,6) f32
    const float* pts = (const float*)d_in[1];   // (2,256,256,2) f32
    float* out = (float*)d_out;                 // (2,256,256,2) f32
    (void)in_sizes; (void)n_in; (void)out_size; (void)d_ws; (void)ws_size;

    dim3 block(256);
    dim3 grid(1024);   // 2 batches * 512 blocks; 8 waves/block * 16 points/wave
    vortex_vel_kernel<<<grid, block, 0, stream>>>(vf, pts, out);
}